// GRU_55911884259481
// MI455X (gfx1250) — compile-verified
//
#include <hip/hip_runtime.h>
#include <math.h>

#define VOCAB 128000
#define EMB   2048
#define HID   4096
#define GATE3 (3 * HID)   // 12288
#define OUT   128000

typedef __attribute__((ext_vector_type(2))) float v2f;
typedef __attribute__((ext_vector_type(8))) float v8f;

// ---------------------------------------------------------------------------
// Wave-level fp32 GEMV over a 16-row tile using V_WMMA_F32_16X16X4_F32.
//   A (16x4, fp32): lanes 0-15 hold rows M=0..15 with K={k,k+1} in the 2 VGPRs,
//                   lanes 16-31 hold K={k+2,k+3}  (ISA 7.12.2, 32-bit A 16x4).
//   B (4x16): broadcast the x chunk into every column -> every column of D is
//             the same 16-row GEMV result.
//   D (16x16 f32): VGPR v, lane-half h -> row M = v + 8*h (columns identical).
// m = lane % 16, g = lane / 16. K must be a multiple of 32.
// ---------------------------------------------------------------------------
__device__ __forceinline__ v8f wave_gemv16(const float* __restrict__ W, int ld,
                                           const float* __restrict__ x, int K,
                                           int m, int g) {
  v8f acc = {};
  const float* wrow = W + (long long)m * ld + 2 * g;
  const float* xp   = x + 2 * g;
  for (int k = 0; k < K; k += 32) {
    // next 128B line of this lane's row (global_prefetch_b8)
    __builtin_prefetch(wrow + k + 32, 0, 0);
#pragma unroll
    for (int kk = k; kk < k + 32; kk += 4) {
      v2f a = *(const v2f*)(wrow + kk);
      v2f b = *(const v2f*)(xp + kk);
      acc = __builtin_amdgcn_wmma_f32_16x16x4_f32(
          /*neg_a=*/false, a, /*neg_b=*/false, b,
          /*c_mod=*/(short)0, acc, /*reuse_a=*/false, /*reuse_b=*/false);
    }
  }
  return acc;
}

// ---------------------------------------------------------------------------
// Kernel 1: gi = W_ih @ x + b_ih ; gh = W_hh @ h + b_hh   (12288 rows each)
// grid = 96 blocks * 8 waves * 16 rows = 12288
// ---------------------------------------------------------------------------
__global__ void gru_gates_kernel(const int* __restrict__ token,
                                 const float* __restrict__ hidden,
                                 const float* __restrict__ W_emb,
                                 const float* __restrict__ W_ih,
                                 const float* __restrict__ W_hh,
                                 const float* __restrict__ b_ih,
                                 const float* __restrict__ b_hh,
                                 float* __restrict__ gi,
                                 float* __restrict__ gh) {
  const int lane = threadIdx.x & 31;
  const int wave = threadIdx.x >> 5;
  const int m = lane & 15;
  const int g = lane >> 4;
  const int row0 = (blockIdx.x * 8 + wave) * 16;

  const float* x = W_emb + (long long)token[0] * EMB;  // embedding row (L2-hot)

  v8f ai = wave_gemv16(W_ih + (long long)row0 * EMB, EMB, x, EMB, m, g);
  v8f ah = wave_gemv16(W_hh + (long long)row0 * HID, HID, hidden, HID, m, g);

  if (m == 0) {  // columns identical; lane-half h owns rows v + 8*h
#pragma unroll
    for (int v = 0; v < 8; ++v) {
      int r = row0 + 8 * g + v;
      gi[r] = ai[v] + b_ih[r];
      gh[r] = ah[v] + b_hh[r];
    }
  }
}

// ---------------------------------------------------------------------------
// Kernel 2: GRU gate math -> h_new  (4096 elems); also writes hidden output.
// ---------------------------------------------------------------------------
__global__ void gru_hnew_kernel(const float* __restrict__ gi,
                                const float* __restrict__ gh,
                                const float* __restrict__ hidden,
                                float* __restrict__ hnew,
                                float* __restrict__ out_hidden) {
  int j = blockIdx.x * blockDim.x + threadIdx.x;
  if (j < HID) {
    float ir = gi[j],           hr = gh[j];
    float iz = gi[HID + j],     hz = gh[HID + j];
    float inn = gi[2 * HID + j], hn = gh[2 * HID + j];
    float r = 1.0f / (1.0f + __expf(-(ir + hr)));
    float z = 1.0f / (1.0f + __expf(-(iz + hz)));
    float n = tanhf(inn + r * hn);
    float h = hidden[j];
    float hv = (1.0f - z) * n + z * h;
    hnew[j] = hv;
    out_hidden[j] = hv;
  }
}

// ---------------------------------------------------------------------------
// Kernel 3: logits = W_dec @ h_new + b_dec (128000 rows) + per-block max.
// grid = 1000 blocks * 8 waves * 16 rows = 128000
// ---------------------------------------------------------------------------
__global__ void gru_decoder_kernel(const float* __restrict__ W_dec,
                                   const float* __restrict__ b_dec,
                                   const float* __restrict__ hnew,
                                   float* __restrict__ logits,
                                   float* __restrict__ partmax) {
  const int lane = threadIdx.x & 31;
  const int m = lane & 15;
  const int g = lane >> 4;
  const int row0 = (blockIdx.x * 8 + (threadIdx.x >> 5)) * 16;

  v8f acc = wave_gemv16(W_dec + (long long)row0 * HID, HID, hnew, HID, m, g);

  float lmax = -3.0e38f;
  if (m == 0) {
#pragma unroll
    for (int v = 0; v < 8; ++v) {
      int r = row0 + 8 * g + v;
      float y = acc[v] + b_dec[r];
      logits[r] = y;
      lmax = fmaxf(lmax, y);
    }
  }

  __shared__ float smax[256];
  smax[threadIdx.x] = lmax;
  __syncthreads();
  for (int s = 128; s > 0; s >>= 1) {
    if (threadIdx.x < s)
      smax[threadIdx.x] = fmaxf(smax[threadIdx.x], smax[threadIdx.x + s]);
    __syncthreads();
  }
  if (threadIdx.x == 0) partmax[blockIdx.x] = smax[0];
}

// ---------------------------------------------------------------------------
// Kernel 4 (1 block, 1024 thr): global max, then sum of exp(logit - max).
// ---------------------------------------------------------------------------
__global__ void gru_softmax_stats_kernel(const float* __restrict__ logits,
                                         const float* __restrict__ partmax,
                                         int nparts,
                                         float* __restrict__ stats) {
  __shared__ float red[1024];
  float lm = -3.0e38f;
  for (int i = threadIdx.x; i < nparts; i += 1024) lm = fmaxf(lm, partmax[i]);
  red[threadIdx.x] = lm;
  __syncthreads();
  for (int s = 512; s > 0; s >>= 1) {
    if (threadIdx.x < s)
      red[threadIdx.x] = fmaxf(red[threadIdx.x], red[threadIdx.x + s]);
    __syncthreads();
  }
  const float gmax = red[0];
  __syncthreads();

  float sum = 0.0f;
  for (int i = threadIdx.x; i < OUT; i += 1024) sum += __expf(logits[i] - gmax);
  red[threadIdx.x] = sum;
  __syncthreads();
  for (int s = 512; s > 0; s >>= 1) {
    if (threadIdx.x < s) red[threadIdx.x] += red[threadIdx.x + s];
    __syncthreads();
  }
  if (threadIdx.x == 0) {
    stats[0] = gmax;
    stats[1] = red[0];
  }
}

// ---------------------------------------------------------------------------
// Kernel 5: out[i] = exp(logits[i] - max) / sum
// ---------------------------------------------------------------------------
__global__ void gru_softmax_out_kernel(const float* __restrict__ logits,
                                       const float* __restrict__ stats,
                                       float* __restrict__ out) {
  int i = blockIdx.x * blockDim.x + threadIdx.x;
  if (i < OUT) out[i] = __expf(logits[i] - stats[0]) * (1.0f / stats[1]);
}

// ---------------------------------------------------------------------------
extern "C" void kernel_launch(void* const* d_in, const int* in_sizes, int n_in,
                              void* d_out, int out_size, void* d_ws, size_t ws_size,
                              hipStream_t stream) {
  const int*   token  = (const int*)  d_in[0];
  const float* hidden = (const float*)d_in[1];
  const float* W_emb  = (const float*)d_in[2];
  const float* W_ih   = (const float*)d_in[3];
  const float* W_hh   = (const float*)d_in[4];
  const float* b_ih   = (const float*)d_in[5];
  const float* b_hh   = (const float*)d_in[6];
  const float* W_dec  = (const float*)d_in[7];
  const float* b_dec  = (const float*)d_in[8];

  float* out = (float*)d_out;          // [0,128000) softmax, [128000,132096) h_new

  // workspace layout (floats)
  float* ws      = (float*)d_ws;
  float* gi      = ws;                          // 12288
  float* gh      = gi + GATE3;                  // 12288
  float* hnew    = gh + GATE3;                  // 4096
  float* logits  = hnew + HID;                  // 128000
  float* partmax = logits + OUT;                // 1000
  float* stats   = partmax + 1024;              // 2

  gru_gates_kernel<<<GATE3 / 128, 256, 0, stream>>>(
      token, hidden, W_emb, W_ih, W_hh, b_ih, b_hh, gi, gh);

  gru_hnew_kernel<<<HID / 256, 256, 0, stream>>>(
      gi, gh, hidden, hnew, out + OUT);

  gru_decoder_kernel<<<OUT / 128, 256, 0, stream>>>(
      W_dec, b_dec, hnew, logits, partmax);

  gru_softmax_stats_kernel<<<1, 1024, 0, stream>>>(
      logits, partmax, OUT / 128, stats);

  gru_softmax_out_kernel<<<(OUT + 255) / 256, 256, 0, stream>>>(
      logits, stats, out);
}